// REN_IQC_gamma_76931454206598
// MI455X (gfx1250) — compile-verified
//
#include <hip/hip_runtime.h>
#include <math.h>

#define TLEN 256

typedef __attribute__((ext_vector_type(2))) float v2f;
typedef __attribute__((ext_vector_type(4))) float v4f;
typedef __attribute__((ext_vector_type(8))) float v8f;

// ---- workspace layout (float offsets) ----
#define WS_FM    0      // 64x64
#define WS_B1    4096   // 64x32
#define WS_EINV  6144   // 64x64
#define WS_C1    10240  // 32x64
#define WS_D11   12288  // 32x32
#define WS_ILAM  13312  // 32
#define WS_D22   13344  // 32x32
#define WS_VECR  14368  // 160x32
#define WS_WR    19488  // 160x32
#define WS_TOTAL 24608

// =====================================================================
// Gauss-Jordan inverse with partial pivoting, aug = N x 2N in LDS,
// right half preinitialized to identity. Block-cooperative.
// =====================================================================
__device__ void gauss_jordan(float* aug, float* fac, int* piv, int N, int tid, int nthr) {
    const int tw = 2 * N;
    for (int k = 0; k < N; ++k) {
        if (tid == 0) {
            int p = k; float best = fabsf(aug[k * tw + k]);
            for (int i = k + 1; i < N; ++i) {
                float v = fabsf(aug[i * tw + k]);
                if (v > best) { best = v; p = i; }
            }
            *piv = p;
        }
        __syncthreads();
        const int p = *piv;
        if (p != k) {
            for (int j = tid; j < tw; j += nthr) {
                float t = aug[k * tw + j];
                aug[k * tw + j] = aug[p * tw + j];
                aug[p * tw + j] = t;
            }
        }
        __syncthreads();
        const float pivinv = 1.0f / aug[k * tw + k];
        __syncthreads();
        for (int j = tid; j < tw; j += nthr) aug[k * tw + j] *= pivinv;
        for (int i = tid; i < N; i += nthr) fac[i] = (i == k) ? 0.0f : aug[i * tw + k];
        __syncthreads();
        for (int idx = tid; idx < N * tw; idx += nthr) {
            const int i = idx / tw, j = idx - i * tw;
            if (i != k) aug[idx] -= fac[i] * aug[k * tw + j];
        }
        __syncthreads();
    }
}

// =====================================================================
// Param kernel A: D22, R_cal^{-1}, vec_r, W_r = vec_r @ R_cal^{-1}
// =====================================================================
__global__ void __launch_bounds__(256)
ren_param_a(const float* __restrict__ gamma, const float* __restrict__ D22L,
            const float* __restrict__ D22U, const float* __restrict__ C2,
            const float* __restrict__ D21,  const float* __restrict__ D12,
            const float* __restrict__ B2,   float* ws) {
    __shared__ float sD22[32 * 32];
    __shared__ float sAug[32 * 64];
    __shared__ float sM[32 * 32];
    __shared__ float sVr[160 * 32];
    __shared__ float sFac[32];
    __shared__ int   sPiv;
    const int tid = threadIdx.x, nthr = blockDim.x;
    const float g = gamma[0];

    // D22 = (g+1)*I + D22_L^T D22_L + (D22_U - D22_U^T)
    for (int idx = tid; idx < 1024; idx += nthr) {
        const int i = idx >> 5, j = idx & 31;
        float s = (i == j) ? (g + 1.0f) : 0.0f;
        for (int k = 0; k < 32; ++k) s += D22L[k * 32 + i] * D22L[k * 32 + j];
        s += D22U[i * 32 + j] - D22U[j * 32 + i];
        sD22[idx] = s;
        ws[WS_D22 + idx] = s;
    }
    __syncthreads();
    // R_cal = -2g*I + D22 + D22^T - 1e-4 * D22^T D22 ; aug right = I
    for (int idx = tid; idx < 1024; idx += nthr) {
        const int i = idx >> 5, j = idx & 31;
        float s = ((i == j) ? (-2.0f * g) : 0.0f) + sD22[i * 32 + j] + sD22[j * 32 + i];
        float q = 0.0f;
        for (int k = 0; k < 32; ++k) q += sD22[k * 32 + i] * sD22[k * 32 + j];
        sAug[i * 64 + j] = s - 1.0e-4f * q;
        sAug[i * 64 + 32 + j] = (i == j) ? 1.0f : 0.0f;
    }
    __syncthreads();
    gauss_jordan(sAug, sFac, &sPiv, 32, tid, nthr);
    // M = I - 1e-4 * D22^T
    for (int idx = tid; idx < 1024; idx += nthr) {
        const int i = idx >> 5, j = idx & 31;
        sM[idx] = ((i == j) ? 1.0f : 0.0f) - 1.0e-4f * sD22[j * 32 + i];
    }
    __syncthreads();
    // vec_r = [ (M C2)^T ; (M D21 - D12^T)^T ; B2 ]  (160 x 32)
    for (int idx = tid; idx < 160 * 32; idx += nthr) {
        const int p = idx >> 5, c = idx & 31;
        float s;
        if (p < 64) {
            s = 0.0f;
            for (int k = 0; k < 32; ++k) s += sM[c * 32 + k] * C2[k * 64 + p];
        } else if (p < 96) {
            const int pp = p - 64;
            s = 0.0f;
            for (int k = 0; k < 32; ++k) s += sM[c * 32 + k] * D21[k * 32 + pp];
            s -= D12[pp * 32 + c];
        } else {
            s = B2[(p - 96) * 32 + c];
        }
        sVr[idx] = s;
        ws[WS_VECR + idx] = s;
    }
    __syncthreads();
    // W_r = vec_r @ R_cal^{-1}
    for (int idx = tid; idx < 160 * 32; idx += nthr) {
        const int i = idx >> 5, c = idx & 31;
        float s = 0.0f;
        for (int k = 0; k < 32; ++k) s += sVr[i * 32 + k] * sAug[k * 64 + 32 + c];
        ws[WS_WR + idx] = s;
    }
}

// =====================================================================
// Param kernel B: H-block extraction -> Fm, B1, C1, D11, invLam, E_inv
// =====================================================================
__device__ __forceinline__ float vq_elem(const float* C2, const float* D21, int i, int k) {
    if (i < 64) return C2[k * 64 + i];
    if (i < 96) return D21[k * 32 + (i - 64)];
    return 0.0f;
}

__device__ float h_elem(const float* __restrict__ X, const float* Wr, const float* Vr,
                        const float* __restrict__ C2, const float* __restrict__ D21,
                        int i, int j) {
    float s = (i == j) ? 1.0e-3f : 0.0f;
    for (int k = 0; k < 160; ++k) s += X[k * 160 + i] * X[k * 160 + j];
    float r = 0.0f;
    for (int k = 0; k < 32; ++k) r += Wr[i * 32 + k] * Vr[j * 32 + k];
    s += r;
    float q = 0.0f;
    for (int k = 0; k < 32; ++k) q += vq_elem(C2, D21, i, k) * vq_elem(C2, D21, j, k);
    return s + 1.0e-4f * q;   // H includes "- psi_q" with Q = -1e-4*I
}

__global__ void __launch_bounds__(256)
ren_param_b(const float* __restrict__ X, const float* __restrict__ Y,
            const float* __restrict__ C2, const float* __restrict__ D21, float* ws) {
    __shared__ float sAug[64 * 128];
    __shared__ float sFac[64];
    __shared__ int   sPiv;
    const int tid = threadIdx.x, nthr = blockDim.x;
    const float* Wr = ws + WS_WR;
    const float* Vr = ws + WS_VECR;

    for (int idx = tid; idx < 64 * 64; idx += nthr) {          // Fm = H31
        const int a = idx >> 6, b = idx & 63;
        ws[WS_FM + idx] = h_elem(X, Wr, Vr, C2, D21, 96 + a, b);
    }
    for (int idx = tid; idx < 64 * 32; idx += nthr) {          // B1 = H32
        const int a = idx >> 5, b = idx & 31;
        ws[WS_B1 + idx] = h_elem(X, Wr, Vr, C2, D21, 96 + a, 64 + b);
    }
    for (int idx = tid; idx < 32 * 64; idx += nthr) {          // C1 = -H21
        const int a = idx >> 6, b = idx & 63;
        ws[WS_C1 + idx] = -h_elem(X, Wr, Vr, C2, D21, 64 + a, b);
    }
    for (int idx = tid; idx < 32 * 32; idx += nthr) {          // D11 strict-lower, invLam
        const int i = idx >> 5, j = idx & 31;
        if (j < i) {
            ws[WS_D11 + idx] = -h_elem(X, Wr, Vr, C2, D21, 64 + i, 64 + j);
        } else if (j == i) {
            const float hv = h_elem(X, Wr, Vr, C2, D21, 64 + i, 64 + j);
            ws[WS_D11 + idx] = 0.0f;
            ws[WS_ILAM + i] = 2.0f / hv;    // invLam = 1/(0.5*H22_ii)
        } else {
            ws[WS_D11 + idx] = 0.0f;
        }
    }
    // E = 0.5*(H11 + H33 + Y - Y^T) into aug left half; right = I
    for (int idx = tid; idx < 64 * 64; idx += nthr) {
        const int a = idx >> 6, b = idx & 63;
        const float e = 0.5f * (h_elem(X, Wr, Vr, C2, D21, a, b) +
                                h_elem(X, Wr, Vr, C2, D21, 96 + a, 96 + b) +
                                Y[a * 64 + b] - Y[b * 64 + a]);
        sAug[a * 128 + b] = e;
        sAug[a * 128 + 64 + b] = (a == b) ? 1.0f : 0.0f;
    }
    __syncthreads();
    gauss_jordan(sAug, sFac, &sPiv, 64, tid, nthr);
    for (int idx = tid; idx < 64 * 64; idx += nthr)
        ws[WS_EINV + idx] = sAug[(idx >> 6) * 128 + 64 + (idx & 63)];
}

// =====================================================================
// Scan kernel: WMMA F32 16x16x4, batch tile = 16 rows per wave32.
//   D(16x16) = W(A-tile 16x4) x dataT(B-tile 4x16), accumulate over K.
//   Weights row-major in LDS; data tiles batch-major [16][feat] in LDS.
// =====================================================================
__device__ __forceinline__ v8f wmma4(v8f c, v2f a, v2f b) {
    return __builtin_amdgcn_wmma_f32_16x16x4_f32(false, a, false, b, (short)0, c, false, false);
}

template<int K, int LDW, int LDD>
__device__ __forceinline__ v8f wmma_tile(v8f c, const float* W, int m0,
                                         const float* D, int lane) {
    const float* wp = W + (m0 + (lane & 15)) * LDW + ((lane >> 4) << 1);
    const float* dp = D + (lane & 15) * LDD + ((lane >> 4) << 1);
#pragma unroll
    for (int k0 = 0; k0 < K; k0 += 4) {
        v2f a = *(const v2f*)(wp + k0);
        v2f b = *(const v2f*)(dp + k0);
        c = wmma4(c, a, b);
    }
    return c;
}

__device__ __forceinline__ void store_cd(float* D, int ldd, int m0, v8f c, int lane) {
    float* p = D + (lane & 15) * ldd + m0 + ((lane >> 4) << 3);
#pragma unroll
    for (int r = 0; r < 8; ++r) p[r] = c[r];
}

__device__ __forceinline__ float tanh_fast(float x) {
    x = fminf(15.0f, fmaxf(-15.0f, x));
    const float e = __expf(2.0f * x);
    return (e - 1.0f) / (e + 1.0f);
}

__global__ void __launch_bounds__(32)
ren_scan(const float* __restrict__ u_in, const float* __restrict__ x0,
         const float* __restrict__ B2g, const float* __restrict__ C2g,
         const float* __restrict__ D12g, const float* __restrict__ D21g,
         const float* __restrict__ ws, float* __restrict__ out) {
    extern __shared__ float smem[];
    float* lFm  = smem;          // 4096
    float* lB1  = smem + 4096;   // 2048
    float* lB2  = smem + 6144;   // 2048
    float* lEin = smem + 8192;   // 4096
    float* lC1  = smem + 12288;  // 2048
    float* lC2  = smem + 14336;  // 2048
    float* lD12 = smem + 16384;  // 1024
    float* lD21 = smem + 17408;  // 1024
    float* lD22 = smem + 18432;  // 1024
    float* lD11 = smem + 19456;  // 1024
    float* lILam= smem + 20480;  // 32
    float* xbuf = smem + 20512;  // 16x64
    float* ubuf = smem + 21536;  // 16x32
    float* wbuf = smem + 22048;  // 16x32
    float* abuf = smem + 22560;  // 16x32
    float* tbuf = smem + 23072;  // 16x64
    float* ybuf = smem + 24096;  // 16x32

    const int lane  = threadIdx.x;
    const int bbase = blockIdx.x * 16;

    for (int i = lane; i < 4096; i += 32) { lFm[i] = ws[WS_FM + i]; lEin[i] = ws[WS_EINV + i]; }
    for (int i = lane; i < 2048; i += 32) {
        lB1[i] = ws[WS_B1 + i]; lC1[i] = ws[WS_C1 + i];
        lB2[i] = B2g[i];        lC2[i] = C2g[i];
    }
    for (int i = lane; i < 1024; i += 32) {
        lD12[i] = D12g[i]; lD21[i] = D21g[i];
        lD22[i] = ws[WS_D22 + i]; lD11[i] = ws[WS_D11 + i];
    }
    lILam[lane] = ws[WS_ILAM + lane];
    for (int i = lane; i < 16 * 64; i += 32) {
        const int r = i >> 6, c = i & 63;
        xbuf[i] = x0[(size_t)(bbase + r) * 64 + c];
    }
    __syncthreads();

    const int b = lane & 15, h = lane >> 4;
    const int urow = lane >> 1, uc0 = (lane & 1) * 16;

    for (int t = 0; t < TLEN; ++t) {
        { // stage u_t tile (coalesced 128B rows)
            const float* src = u_in + ((size_t)(bbase + urow) * TLEN + t) * 32 + uc0;
            float* dst = ubuf + urow * 32 + uc0;
            *(v4f*)(dst)      = *(const v4f*)(src);
            *(v4f*)(dst + 4)  = *(const v4f*)(src + 4);
            *(v4f*)(dst + 8)  = *(const v4f*)(src + 8);
            *(v4f*)(dst + 12) = *(const v4f*)(src + 12);
        }
        // a^T = C1 x^T + D12 u^T
#pragma unroll
        for (int mt = 0; mt < 2; ++mt) {
            v8f c = {};
            c = wmma_tile<64, 64, 64>(c, lC1, mt * 16, xbuf, lane);
            c = wmma_tile<32, 32, 32>(c, lD12, mt * 16, ubuf, lane);
            store_cd(abuf, 32, mt * 16, c, lane);
        }
        // sequential tanh forward-substitution (D11 strictly lower triangular)
        float wreg[16];
#pragma unroll
        for (int r = 0; r < 16; ++r) wreg[r] = 0.0f;
#pragma unroll
        for (int i = 0; i < 32; ++i) {
            const float* drow = lD11 + i * 32 + h * 16;
            float part = 0.0f;
#pragma unroll
            for (int r = 0; r < 16; ++r) part += wreg[r] * drow[r];
            const float vsum = part + __shfl_xor(part, 16, 32);
            const float wv = tanh_fast((abuf[b * 32 + i] + vsum) * lILam[i]);
            if (h == (i >> 4)) wreg[i & 15] = wv;
        }
#pragma unroll
        for (int r = 0; r < 16; ++r) wbuf[b * 32 + h * 16 + r] = wreg[r];
        // tmp^T = Fm x^T + B1 w^T + B2 u^T
#pragma unroll
        for (int mt = 0; mt < 4; ++mt) {
            v8f c = {};
            c = wmma_tile<64, 64, 64>(c, lFm, mt * 16, xbuf, lane);
            c = wmma_tile<32, 32, 32>(c, lB1, mt * 16, wbuf, lane);
            c = wmma_tile<32, 32, 32>(c, lB2, mt * 16, ubuf, lane);
            store_cd(tbuf, 64, mt * 16, c, lane);
        }
        // x_new^T = E_inv tmp^T
#pragma unroll
        for (int mt = 0; mt < 4; ++mt) {
            v8f c = {};
            c = wmma_tile<64, 64, 64>(c, lEin, mt * 16, tbuf, lane);
            store_cd(xbuf, 64, mt * 16, c, lane);
        }
        // y^T = C2 x_new^T + D21 w^T + D22 u^T
#pragma unroll
        for (int mt = 0; mt < 2; ++mt) {
            v8f c = {};
            c = wmma_tile<64, 64, 64>(c, lC2, mt * 16, xbuf, lane);
            c = wmma_tile<32, 32, 32>(c, lD21, mt * 16, wbuf, lane);
            c = wmma_tile<32, 32, 32>(c, lD22, mt * 16, ubuf, lane);
            store_cd(ybuf, 32, mt * 16, c, lane);
        }
        { // write y tile (coalesced 128B rows)
            float* dst = out + ((size_t)(bbase + urow) * TLEN + t) * 32 + uc0;
            const float* src = ybuf + urow * 32 + uc0;
            *(v4f*)(dst)      = *(const v4f*)(src);
            *(v4f*)(dst + 4)  = *(const v4f*)(src + 4);
            *(v4f*)(dst + 8)  = *(const v4f*)(src + 8);
            *(v4f*)(dst + 12) = *(const v4f*)(src + 12);
        }
    }
}

extern "C" void kernel_launch(void* const* d_in, const int* in_sizes, int n_in,
                              void* d_out, int out_size, void* d_ws, size_t ws_size,
                              hipStream_t stream) {
    const float* u_in  = (const float*)d_in[0];
    const float* x0    = (const float*)d_in[1];
    const float* X     = (const float*)d_in[2];
    const float* Y     = (const float*)d_in[3];
    const float* B2    = (const float*)d_in[4];
    const float* C2    = (const float*)d_in[5];
    const float* D12   = (const float*)d_in[6];
    const float* D22L  = (const float*)d_in[7];
    const float* D22U  = (const float*)d_in[8];
    const float* D21   = (const float*)d_in[9];
    const float* gamma = (const float*)d_in[10];
    float* ws  = (float*)d_ws;
    float* out = (float*)d_out;

    ren_param_a<<<1, 256, 0, stream>>>(gamma, D22L, D22U, C2, D21, D12, B2, ws);
    ren_param_b<<<1, 256, 0, stream>>>(X, Y, C2, D21, ws);

    const size_t lds_bytes = (size_t)WS_TOTAL * sizeof(float); // 98.4 KB dynamic LDS (<320KB WGP)
    ren_scan<<<128, 32, lds_bytes, stream>>>(u_in, x0, B2, C2, D12, D21, ws, out);
}